// NeighborList_55611236549210
// MI455X (gfx1250) — compile-verified
//
#include <hip/hip_runtime.h>
#include <hip/hip_bf16.h>
#include <stdint.h>

typedef __attribute__((ext_vector_type(2))) float v2f;
typedef __attribute__((ext_vector_type(8))) float v8f;

#define RADIUS 0.1f

// Block = 32 rows x 1024 neighbor-cols (= 32 packed words per row).
// Wave w (of 8) computes a 32-row x 128-col sub-tile: 16 x v_wmma_f32_16x16x4_f32.
//
// WMMA formulation (K=4):
//   A row i = (-2x_i, -2y_i, -2z_i, |p_i|^2),  B col k = (x_k, y_k, z_k, 1)
//   D[i,k]  = |p_i|^2 - 2 p_i . p_k ;  neighbor test: D[i,k] <= r^2 - |p_k|^2.
//   The threshold is per-COLUMN and the C/D layout's column index == lane&15 ==
//   the B-operand lane mapping, so it is per-lane with no cross-lane traffic.
//
// Bit packing: for accumulator VGPR v, a wave32 ballot of (D <= thr) yields
//   bits 0-15  -> row (base+v),   16 consecutive neighbor bits
//   bits 16-31 -> row (base+v+8), same neighbor bits
// Two adjacent 16-col tiles splice into full 32-bit words with 3 SALU ops.
// Words are staged in LDS so global stores are 128B-coalesced uint4 writes.
__global__ __launch_bounds__(256)
void neighborlist_wmma_kernel(const float* __restrict__ locs,
                              unsigned int* __restrict__ out,
                              int N) {
    __shared__ uint4 smem4[256];          // [row 0..31][wordgroup 0..7] (4KB)

    const int nwords = N >> 5;
    const int lane = threadIdx.x;         // 0..31
    const int wave = threadIdx.y;         // 0..7
    const int tid  = wave * 32 + lane;
    const bool hi  = (lane & 16) != 0;    // upper half-wave
    const int l15  = lane & 15;

    const int b    = blockIdx.z;
    const int row0 = blockIdx.y * 32;     // 32-row block base
    const int kw0  = blockIdx.x * 32;     // first packed-word index of block

    const float* __restrict__ P = locs + (size_t)b * N * 3;
    const float r2 = RADIUS * RADIUS;

    // ---- A operand: two 16-row tiles, loaded once, reused by all 16 WMMAs ----
    const int ra = row0 + l15;
    const int rb = row0 + 16 + l15;
    const float ax = P[ra * 3 + 0], ay = P[ra * 3 + 1], az = P[ra * 3 + 2];
    const float bx = P[rb * 3 + 0], by = P[rb * 3 + 1], bz = P[rb * 3 + 2];
    const float asq = ax * ax + ay * ay + az * az;
    const float bsq = bx * bx + by * by + bz * bz;
    v2f aLo, aHi;
    aLo.x = hi ? (-2.0f * az) : (-2.0f * ax);
    aLo.y = hi ? asq          : (-2.0f * ay);
    aHi.x = hi ? (-2.0f * bz) : (-2.0f * bx);
    aHi.y = hi ? bsq          : (-2.0f * by);

    const v8f zacc = {0.f, 0.f, 0.f, 0.f, 0.f, 0.f, 0.f, 0.f};

    unsigned int wsel[4];
    #pragma unroll
    for (int j = 0; j < 4; ++j) {
        const int kw = kw0 + wave * 4 + j;   // this word-column
        const int c0 = kw * 32;

        // ---- B operand: two 16-col tiles ----
        const int ca = c0 + l15;
        const int cb = c0 + 16 + l15;
        const float cx0 = P[ca * 3 + 0], cy0 = P[ca * 3 + 1], cz0 = P[ca * 3 + 2];
        const float cx1 = P[cb * 3 + 0], cy1 = P[cb * 3 + 1], cz1 = P[cb * 3 + 2];
        v2f bL, bR;
        bL.x = hi ? cz0 : cx0;  bL.y = hi ? 1.0f : cy0;
        bR.x = hi ? cz1 : cx1;  bR.y = hi ? 1.0f : cy1;
        const float thrL = r2 - (cx0 * cx0 + cy0 * cy0 + cz0 * cz0);
        const float thrR = r2 - (cx1 * cx1 + cy1 * cy1 + cz1 * cz1);

        // 8 args: (neg_a, A, neg_b, B, c_mod, C, reuse_a, reuse_b)
        v8f cLL = __builtin_amdgcn_wmma_f32_16x16x4_f32(false, aLo, false, bL,
                                                        (short)0, zacc, false, false);
        v8f cLR = __builtin_amdgcn_wmma_f32_16x16x4_f32(false, aLo, false, bR,
                                                        (short)0, zacc, false, false);
        v8f cHL = __builtin_amdgcn_wmma_f32_16x16x4_f32(false, aHi, false, bL,
                                                        (short)0, zacc, false, false);
        v8f cHR = __builtin_amdgcn_wmma_f32_16x16x4_f32(false, aHi, false, bR,
                                                        (short)0, zacc, false, false);

        unsigned int w = 0u;
        #pragma unroll
        for (int v = 0; v < 8; ++v) {
            const unsigned mLL = __builtin_amdgcn_ballot_w32(cLL[v] <= thrL);
            const unsigned mLR = __builtin_amdgcn_ballot_w32(cLR[v] <= thrR);
            const unsigned mHL = __builtin_amdgcn_ballot_w32(cHL[v] <= thrL);
            const unsigned mHR = __builtin_amdgcn_ballot_w32(cHR[v] <= thrR);

            const unsigned wLlo = (mLL & 0xFFFFu) | (mLR << 16);          // row row0+v
            const unsigned wLhi = (mLL >> 16)     | (mLR & 0xFFFF0000u);  // row row0+v+8
            const unsigned wHlo = (mHL & 0xFFFFu) | (mHR << 16);          // row row0+16+v
            const unsigned wHhi = (mHL >> 16)     | (mHR & 0xFFFF0000u);  // row row0+24+v

            // lane==r predicates are shared across j (CSE into SGPR masks)
            w = (lane == v)      ? wLlo : w;
            w = (lane == v + 8)  ? wLhi : w;
            w = (lane == v + 16) ? wHlo : w;
            w = (lane == v + 24) ? wHhi : w;
        }
        wsel[j] = w;
    }

    // ---- stage to LDS: lane holds 4 consecutive words of row (row0+lane) ----
    uint4 pack;
    pack.x = wsel[0]; pack.y = wsel[1]; pack.z = wsel[2]; pack.w = wsel[3];
    smem4[lane * 8 + wave] = pack;        // ds_store_b128
    __syncthreads();

    // ---- coalesced writeback: 8 threads per row cover 32 words = 128B line ----
    const int orow = tid >> 3;            // 0..31
    const int og   = tid & 7;             // 0..7 word-group
    const uint4 vout = smem4[orow * 8 + og];
    uint4* dst = (uint4*)(out + ((size_t)b * N + row0 + orow) * nwords + kw0);
    dst[og] = vout;                       // global_store_b128
}

extern "C" void kernel_launch(void* const* d_in, const int* in_sizes, int n_in,
                              void* d_out, int out_size, void* d_ws, size_t ws_size,
                              hipStream_t stream) {
    (void)n_in; (void)d_ws; (void)ws_size; (void)out_size;
    const float* locs = (const float*)d_in[0];
    unsigned int* out = (unsigned int*)d_out;  // raw uint32 words (reference packs bits)

    const int B = 4;
    const int N = in_sizes[0] / (B * 3);       // 8192

    // grid: x = 32-word groups (N/1024), y = 32-row groups (N/32), z = batch
    dim3 block(32, 8, 1);
    dim3 grid(N / 1024, N / 32, B);
    neighborlist_wmma_kernel<<<grid, block, 0, stream>>>(locs, out, N);
}